// HomogeneousGatNodeModule_79826262163706
// MI455X (gfx1250) — compile-verified
//
#include <hip/hip_runtime.h>
#include <math.h>

#define NN 100000
#define EE 600000
#define DD 128
#define HH 4
#define CC 32
#define NEG_SLOPE 0.2f

typedef __attribute__((ext_vector_type(2))) float v2f;
typedef __attribute__((ext_vector_type(8))) float v8f;

// ---------- helpers ----------

__device__ __forceinline__ void atomicMaxF(float* addr, float val) {
    // sign-split trick: correct total order on float bits
    if (val >= 0.0f) {
        atomicMax((int*)addr, __float_as_int(val));
    } else {
        atomicMin((unsigned int*)addr, __float_as_uint(val));
    }
}

// ---------- init kernels ----------

__global__ void init_out_kernel(float* __restrict__ out, const float* __restrict__ bias) {
    long long i = (long long)blockIdx.x * blockDim.x + threadIdx.x;
    if (i < (long long)NN * DD) out[i] = bias[i & (DD - 1)];
}

__global__ void init_nh_kernel(float* __restrict__ amax, float* __restrict__ denom) {
    int i = blockIdx.x * blockDim.x + threadIdx.x;
    if (i < NN * HH) {
        amax[i]  = -__builtin_inff();
        denom[i] = 0.0f;
    }
}

// ---------- fused dual GEMM: x_l = x@W_l + b_l ; x_r = x@W_r + b_r ----------
// One wave per 16x16 output tile; shared A fragments, two accumulators.

__global__ void gemm_nodes_wmma(const float* __restrict__ x,
                                const float* __restrict__ Wl, const float* __restrict__ bl,
                                const float* __restrict__ Wr, const float* __restrict__ br,
                                float* __restrict__ xl, float* __restrict__ xr) {
    const int wave = (blockIdx.x * blockDim.x + threadIdx.x) >> 5;
    const int lane = threadIdx.x & 31;
    const int tiles = (NN / 16) * (DD / 16);
    if (wave >= tiles) return;                 // wave-uniform: EXEC stays all-ones

    const int rowTile = wave >> 3;             // DD/16 == 8 column tiles
    const int colTile = wave & 7;
    const int m   = lane & 15;
    const int klo = (lane >> 4) * 2;           // K offset within 4-wide step
    const long long arow = (long long)(rowTile * 16 + m) * DD;
    const int col = colTile * 16 + m;

    v8f accL = {};
    v8f accR = {};
    for (int k = 0; k < DD; k += 4) {
        v2f a, bL, bR;
        a.x  = x[arow + k + klo];
        a.y  = x[arow + k + klo + 1];
        bL.x = Wl[(long long)(k + klo)     * DD + col];
        bL.y = Wl[(long long)(k + klo + 1) * DD + col];
        bR.x = Wr[(long long)(k + klo)     * DD + col];
        bR.y = Wr[(long long)(k + klo + 1) * DD + col];
        accL = __builtin_amdgcn_wmma_f32_16x16x4_f32(false, a, false, bL, (short)0, accL, false, false);
        accR = __builtin_amdgcn_wmma_f32_16x16x4_f32(false, a, false, bR, (short)0, accR, false, false);
    }

    const int rbase = rowTile * 16 + ((lane < 16) ? 0 : 8);
    const float addL = bl[col];
    const float addR = br[col];
#pragma unroll
    for (int j = 0; j < 8; ++j) {
        xl[(long long)(rbase + j) * DD + col] = accL[j] + addL;
        xr[(long long)(rbase + j) * DD + col] = accR[j] + addR;
    }
}

// ---------- fused edge GEMM + logits ----------
// Each wave: 16 edges. WMMA computes e = edge_attr[tile]@W_e into LDS
// (never hits HBM), then lanes finish leaky_relu(x_l[src]+x_r[dst]+e)·att
// per head and atomic-max the segment max.

__global__ void __launch_bounds__(128)
edge_logits_wmma(const float* __restrict__ ea, const float* __restrict__ We,
                 const int* __restrict__ src, const int* __restrict__ dst,
                 const float* __restrict__ xl, const float* __restrict__ xr,
                 const float* __restrict__ att,
                 float* __restrict__ logits, float* __restrict__ amax) {
    __shared__ float smem[4 * 16 * DD];        // 4 waves/block * 16 edges * 128 cols = 32KB
    const int wv   = threadIdx.x >> 5;
    const int lane = threadIdx.x & 31;
    const int tile = blockIdx.x * 4 + wv;
    float* lds = smem + wv * 16 * DD;

    if (tile < EE / 16) {                      // wave-uniform guard
        const int base = tile * 16;
        const int m    = lane & 15;
        const int klo  = (lane >> 4) * 2;
        const long long arow = (long long)(base + m) * DD;   // A: M = lane&15 -> edge row
        const int ldrow = (lane < 16) ? 0 : 8;

        for (int t = 0; t < DD / 16; ++t) {
            const int col = t * 16 + m;
            v8f acc = {};
            for (int k = 0; k < DD; k += 4) {
                v2f a, b;
                a.x = ea[arow + k + klo];
                a.y = ea[arow + k + klo + 1];
                b.x = We[(long long)(k + klo)     * DD + col];
                b.y = We[(long long)(k + klo + 1) * DD + col];
                acc = __builtin_amdgcn_wmma_f32_16x16x4_f32(false, a, false, b, (short)0, acc, false, false);
            }
#pragma unroll
            for (int j = 0; j < 8; ++j)
                lds[(ldrow + j) * DD + col] = acc[j];
        }
    }
    __syncthreads();                           // uniform across block

    if (tile < EE / 16) {
        // lane handles edge (lane&15), column half (lane>>4)*64 -> two heads
        const int e_s = tile * 16 + (lane & 15);
        const int s   = src[e_s];
        const int d   = dst[e_s];
        const int half  = lane >> 4;
        const int cbase = half * 64;
        const float* ep  = lds + (lane & 15) * DD;
        const float* xlp = xl + (long long)s * DD;
        const float* xrp = xr + (long long)d * DD;
        float p0 = 0.0f, p1 = 0.0f;
#pragma unroll 4
        for (int cc = 0; cc < 64; ++cc) {
            const int col = cbase + cc;
            float v = ep[col] + xlp[col] + xrp[col];
            v = (v > 0.0f) ? v : v * NEG_SLOPE;
            const float w = v * att[col];      // att flat [H*C] == [128]
            if (cc < CC) p0 += w; else p1 += w;
        }
        const int h0 = half * 2;
        logits[(long long)e_s * HH + h0]     = p0;
        logits[(long long)e_s * HH + h0 + 1] = p1;
        atomicMaxF(&amax[d * HH + h0],     p0);
        atomicMaxF(&amax[d * HH + h0 + 1], p1);
    }
}

// ---------- softmax pieces ----------

__global__ void amax_fix_kernel(float* __restrict__ amax) {
    int i = blockIdx.x * blockDim.x + threadIdx.x;
    if (i < NN * HH) {
        float v = amax[i];
        amax[i] = isfinite(v) ? v : 0.0f;
    }
}

__global__ void edge_exp_kernel(float* __restrict__ lg, const float* __restrict__ amax,
                                float* __restrict__ denom, const int* __restrict__ dst) {
    long long i = (long long)blockIdx.x * blockDim.x + threadIdx.x;
    if (i < (long long)EE * HH) {
        const int e = (int)(i >> 2);           // HH == 4
        const int h = (int)(i & 3);
        const int d = dst[e];
        const float ex = __expf(lg[i] - amax[d * HH + h]);
        lg[i] = ex;                            // reuse buffer as ex
        atomicAdd(&denom[d * HH + h], ex);
    }
}

// ---------- alpha-weighted scatter aggregation ----------

__global__ void aggregate_kernel(const float* __restrict__ ex, const float* __restrict__ denom,
                                 const float* __restrict__ xl,
                                 const int* __restrict__ src, const int* __restrict__ dst,
                                 float* __restrict__ out) {
    long long i = (long long)blockIdx.x * blockDim.x + threadIdx.x;
    if (i < (long long)EE * DD) {
        const int e   = (int)(i >> 7);         // DD == 128
        const int col = (int)(i & 127);
        const int h   = col >> 5;              // CC == 32
        const int s   = src[e];
        const int d   = dst[e];
        const float alpha = ex[(long long)e * HH + h] / (denom[d * HH + h] + 1e-16f);
        atomicAdd(&out[(long long)d * DD + col], alpha * xl[(long long)s * DD + col]);
    }
}

// ---------- launch ----------

extern "C" void kernel_launch(void* const* d_in, const int* in_sizes, int n_in,
                              void* d_out, int out_size, void* d_ws, size_t ws_size,
                              hipStream_t stream) {
    const float* x    = (const float*)d_in[0];
    const int*   ei   = (const int*)  d_in[1];
    const float* ea   = (const float*)d_in[2];
    const float* Wl   = (const float*)d_in[3];
    const float* bl   = (const float*)d_in[4];
    const float* Wr   = (const float*)d_in[5];
    const float* br   = (const float*)d_in[6];
    const float* We   = (const float*)d_in[7];
    const float* att  = (const float*)d_in[8];
    const float* bias = (const float*)d_in[9];
    float* out = (float*)d_out;

    const int* src = ei;
    const int* dst = ei + EE;

    // workspace layout (floats): xl[N*128] xr[N*128] logits/ex[E*4] amax[N*4] denom[N*4]
    float* xl    = (float*)d_ws;
    float* xr    = xl + (size_t)NN * DD;
    float* lg    = xr + (size_t)NN * DD;
    float* amax  = lg + (size_t)EE * HH;
    float* denom = amax + (size_t)NN * HH;

    // init
    {
        long long n = (long long)NN * DD;
        init_out_kernel<<<(unsigned)((n + 255) / 256), 256, 0, stream>>>(out, bias);
        init_nh_kernel<<<(NN * HH + 255) / 256, 256, 0, stream>>>(amax, denom);
    }

    // dual node GEMM: (N/16)*(128/16) waves, 8 waves per 256-thread block
    {
        const int waves = (NN / 16) * (DD / 16);      // 50000
        gemm_nodes_wmma<<<(waves + 7) / 8, 256, 0, stream>>>(x, Wl, bl, Wr, br, xl, xr);
    }

    // fused edge GEMM + logits + segment max: E/16 waves, 4 waves/block
    {
        const int tiles = EE / 16;                    // 37500
        edge_logits_wmma<<<(tiles + 3) / 4, 128, 0, stream>>>(ea, We, src, dst, xl, xr, att, lg, amax);
    }

    amax_fix_kernel<<<(NN * HH + 255) / 256, 256, 0, stream>>>(amax);

    {
        long long n = (long long)EE * HH;
        edge_exp_kernel<<<(unsigned)((n + 255) / 256), 256, 0, stream>>>(lg, amax, denom, dst);
    }

    {
        long long n = (long long)EE * DD;
        aggregate_kernel<<<(unsigned)((n + 255) / 256), 256, 0, stream>>>(lg, denom, xl, src, dst, out);
    }
}